// _CCNonLocalNd_69063074120029
// MI455X (gfx1250) — compile-verified
//
#include <hip/hip_runtime.h>

#define NSP (64 * 64 * 64) // 262144 spatial voxels

typedef __attribute__((ext_vector_type(2))) float v2f;
typedef __attribute__((ext_vector_type(8))) float v8f;

__device__ __forceinline__ v8f wmma_f32_4(v2f a, v2f b, v8f c) {
    // D = A(16x4, f32) * B(4x16, f32) + C(16x16, f32)
    return __builtin_amdgcn_wmma_f32_16x16x4_f32(false, a, false, b, (short)0, c,
                                                 false, false);
}

// ---------------------------------------------------------------------------
// Kernel 1: projections.  q,k = [theta;phi](16x64) @ X, v = g_w(64x64) @ X
// One block per 64 contiguous spatial positions.
// ---------------------------------------------------------------------------
__global__ __launch_bounds__(128) void ccnl_proj_kernel(
    const float* __restrict__ x, const float* __restrict__ theta_w,
    const float* __restrict__ theta_b, const float* __restrict__ phi_w,
    const float* __restrict__ phi_b, const float* __restrict__ g_w,
    const float* __restrict__ g_b, float* __restrict__ q, float* __restrict__ k,
    float* __restrict__ v) {
    __shared__ float Xs[64][64];   // [c][pos]
    __shared__ float Wqk[16][65];  // rows 0-7 theta, 8-15 phi (padded)
    __shared__ float Wv[64][65];   // g_w (padded)

    const int t = threadIdx.x;
    const int pos0 = blockIdx.x * 64;

    for (int idx = t; idx < 4096; idx += 128) {
        int c = idx >> 6, p = idx & 63;
        Xs[c][p] = x[c * NSP + pos0 + p];
    }
    for (int idx = t; idx < 1024; idx += 128) {
        int o = idx >> 6, c = idx & 63;
        Wqk[o][c] = (o < 8) ? theta_w[o * 64 + c] : phi_w[(o - 8) * 64 + c];
    }
    for (int idx = t; idx < 4096; idx += 128) {
        int o = idx >> 6, c = idx & 63;
        Wv[o][c] = g_w[o * 64 + c];
    }
    __syncthreads();

    const int wave = t >> 5, lane = t & 31;
    const int m = lane & 15, half = lane >> 4;

    // ---- q/k: one 16x16 output tile per wave (cols 16*wave..) ----
    {
        v8f acc = {0.f, 0.f, 0.f, 0.f, 0.f, 0.f, 0.f, 0.f};
        const int n = 16 * wave + m;
#pragma unroll
        for (int ks = 0; ks < 16; ++ks) {
            const int kb = 4 * ks + 2 * half;
            v2f a = {Wqk[m][kb], Wqk[m][kb + 1]};
            v2f b = {Xs[kb][n], Xs[kb + 1][n]};
            acc = wmma_f32_4(a, b, acc);
        }
        const int pos = pos0 + n;
#pragma unroll
        for (int r = 0; r < 8; ++r) {
            const int ch = r + 8 * half;  // D row = output channel
            if (ch < 8)
                q[ch * NSP + pos] = acc[r] + theta_b[ch];
            else
                k[(ch - 8) * NSP + pos] = acc[r] + phi_b[ch - 8];
        }
    }

    // ---- v: 16-row strip per wave, 4 col tiles ----
    {
        v8f acc[4];
#pragma unroll
        for (int tn = 0; tn < 4; ++tn) acc[tn] = (v8f){0.f, 0.f, 0.f, 0.f, 0.f, 0.f, 0.f, 0.f};
        const int arow = 16 * wave + m;
#pragma unroll
        for (int ks = 0; ks < 16; ++ks) {
            const int kb = 4 * ks + 2 * half;
            v2f a = {Wv[arow][kb], Wv[arow][kb + 1]};
#pragma unroll
            for (int tn = 0; tn < 4; ++tn) {
                const int n = 16 * tn + m;
                v2f b = {Xs[kb][n], Xs[kb + 1][n]};
                acc[tn] = wmma_f32_4(a, b, acc[tn]);
            }
        }
#pragma unroll
        for (int tn = 0; tn < 4; ++tn) {
#pragma unroll
            for (int r = 0; r < 8; ++r) {
                const int co = 16 * wave + r + 8 * half;
                const int pos = pos0 + 16 * tn + m;
                v[co * NSP + pos] = acc[tn][r] + g_b[co];
            }
        }
    }
}

// ---------------------------------------------------------------------------
// Kernel 2: per-voxel softmax statistics over all 192 scores (eH/eW/eD).
// One block per (h,w), thread t = d.  Online max/sum-exp.
// ---------------------------------------------------------------------------
__global__ __launch_bounds__(64) void ccnl_stats_kernel(
    const float* __restrict__ q, const float* __restrict__ k,
    float* __restrict__ m_out, float* __restrict__ zinv_out) {
    __shared__ float kD[8][64];  // k line along D at (h,w)

    const int t = threadIdx.x;  // d coordinate
    const int h = blockIdx.x >> 6, w = blockIdx.x & 63;
    const int sline = h * 4096 + w * 64;

    for (int idx = t; idx < 512; idx += 64) {
        int c = idx >> 6, g = idx & 63;
        kD[c][g] = k[c * NSP + sline + g];
    }
    float qv[8];
#pragma unroll
    for (int c = 0; c < 8; ++c) qv[c] = q[c * NSP + sline + t];
    __syncthreads();

    float mr = -INFINITY, Z = 0.f;
    // eD (no mask)
    for (int g = 0; g < 64; ++g) {
        float e = 0.f;
#pragma unroll
        for (int c = 0; c < 8; ++c) e += qv[c] * kD[c][g];
        float mn = fmaxf(mr, e);
        Z = Z * __expf(mr - mn) + __expf(e - mn);
        mr = mn;
    }
    // eH (diagonal g==h masked)
    for (int g = 0; g < 64; ++g) {
        if (g == h) continue;
        float e = 0.f;
#pragma unroll
        for (int c = 0; c < 8; ++c) e += qv[c] * k[c * NSP + g * 4096 + w * 64 + t];
        float mn = fmaxf(mr, e);
        Z = Z * __expf(mr - mn) + __expf(e - mn);
        mr = mn;
    }
    // eW (no mask)
    for (int g = 0; g < 64; ++g) {
        float e = 0.f;
#pragma unroll
        for (int c = 0; c < 8; ++c) e += qv[c] * k[c * NSP + h * 4096 + g * 64 + t];
        float mn = fmaxf(mr, e);
        Z = Z * __expf(mr - mn) + __expf(e - mn);
        mr = mn;
    }
    const int s = sline + t;
    m_out[s] = mr;
    zinv_out[s] = 1.f / Z;
}

// ---------------------------------------------------------------------------
// Kernels 3-5: per-axis attention aggregation (parametrized by line stride).
// Block handles one 64-voxel line.  attn(64x64) built in LDS, then
// O(64 pos x 64 ch) = attn @ v_line^T via WMMA f32 16x16x4.
// ---------------------------------------------------------------------------
__global__ __launch_bounds__(128) void ccnl_axis_kernel(
    const float* __restrict__ q, const float* __restrict__ k,
    const float* __restrict__ v, const float* __restrict__ m_ws,
    const float* __restrict__ zinv_ws, const float* __restrict__ x,
    const float* __restrict__ gamma, float* __restrict__ out, int strideA,
    int M1, int M2, int addX, int maskDiag) {
    __shared__ float qL[8][64];    // [c][i]
    __shared__ float kL[8][64];    // [c][g]
    __shared__ float vT[64][65];   // [g][c] (padded)
    __shared__ float attn[64][65]; // attn[i][g]; reused as out_s[c][i]
    __shared__ float mL[64];
    __shared__ float zL[64];

    const int t = threadIdx.x;
    const int sbase = (blockIdx.x >> 6) * M1 + (blockIdx.x & 63) * M2;

    for (int idx = t; idx < 1024; idx += 128) {
        int c = (idx >> 6) & 7, g = idx & 63;
        if (idx < 512)
            qL[c][g] = q[c * NSP + sbase + g * strideA];
        else
            kL[c][g] = k[c * NSP + sbase + g * strideA];
    }
    for (int idx = t; idx < 4096; idx += 128) {
        int c = idx >> 6, g = idx & 63;
        vT[g][c] = v[c * NSP + sbase + g * strideA];
    }
    if (t < 64) {
        mL[t] = m_ws[sbase + t * strideA];
        zL[t] = zinv_ws[sbase + t * strideA];
    }
    __syncthreads();

    // scores + softmax weights (VALU: only 8 MACs per element)
    for (int idx = t; idx < 4096; idx += 128) {
        int i = idx >> 6, g = idx & 63;
        float a;
        if (maskDiag && i == g) {
            a = 0.f;
        } else {
            float e = 0.f;
#pragma unroll
            for (int c = 0; c < 8; ++c) e += qL[c][i] * kL[c][g];
            a = __expf(e - mL[i]) * zL[i];
        }
        attn[i][g] = a;
    }
    __syncthreads();

    // O[i][ch] = sum_g attn[i][g] * vT[g][ch]  (64x64x64 GEMM, WMMA)
    const int wave = t >> 5, lane = t & 31;
    const int m = lane & 15, half = lane >> 4;
    v8f acc[4];
#pragma unroll
    for (int tn = 0; tn < 4; ++tn) acc[tn] = (v8f){0.f, 0.f, 0.f, 0.f, 0.f, 0.f, 0.f, 0.f};
    const int arow = 16 * wave + m;
#pragma unroll
    for (int ks = 0; ks < 16; ++ks) {
        const int kb = 4 * ks + 2 * half;
        v2f a = {attn[arow][kb], attn[arow][kb + 1]};
#pragma unroll
        for (int tn = 0; tn < 4; ++tn) {
            const int n = 16 * tn + m;
            v2f b = {vT[kb][n], vT[kb + 1][n]};
            acc[tn] = wmma_f32_4(a, b, acc[tn]);
        }
    }
    __syncthreads();  // all attn reads done; reuse buffer as out_s[c][i]
#pragma unroll
    for (int tn = 0; tn < 4; ++tn) {
#pragma unroll
        for (int r = 0; r < 8; ++r) {
            const int c = 16 * tn + m;              // D col = channel
            const int i = 16 * wave + r + 8 * half; // D row = line position
            attn[c][i] = acc[tn][r];
        }
    }
    __syncthreads();

    const float gm = gamma[0];
    for (int idx = t; idx < 4096; idx += 128) {
        int c = idx >> 6, i = idx & 63;
        const int addr = c * NSP + sbase + i * strideA;
        const float val = gm * attn[c][i];
        if (addX)
            out[addr] = x[addr] + val;
        else
            out[addr] += val;
    }
}

// ---------------------------------------------------------------------------
extern "C" void kernel_launch(void* const* d_in, const int* in_sizes, int n_in,
                              void* d_out, int out_size, void* d_ws,
                              size_t ws_size, hipStream_t stream) {
    const float* x = (const float*)d_in[0];
    const float* theta_w = (const float*)d_in[1];
    const float* theta_b = (const float*)d_in[2];
    const float* phi_w = (const float*)d_in[3];
    const float* phi_b = (const float*)d_in[4];
    const float* g_w = (const float*)d_in[5];
    const float* g_b = (const float*)d_in[6];
    const float* gamma = (const float*)d_in[7];
    float* out = (float*)d_out;

    float* ws = (float*)d_ws;
    float* q = ws;                 //  8*NSP
    float* k = ws + 8l * NSP;      //  8*NSP
    float* v = ws + 16l * NSP;     // 64*NSP
    float* m = ws + 80l * NSP;     //  1*NSP
    float* z = ws + 81l * NSP;     //  1*NSP   (total 82*NSP floats ~ 86 MB)

    ccnl_proj_kernel<<<4096, 128, 0, stream>>>(x, theta_w, theta_b, phi_w,
                                               phi_b, g_w, g_b, q, k, v);
    ccnl_stats_kernel<<<4096, 64, 0, stream>>>(q, k, m, z);
    // axis D: block = (h,w), line stride 1, writes out = x + gamma*oD
    ccnl_axis_kernel<<<4096, 128, 0, stream>>>(q, k, v, m, z, x, gamma, out,
                                               /*strideA=*/1, /*M1=*/4096,
                                               /*M2=*/64, /*addX=*/1,
                                               /*maskDiag=*/0);
    // axis W: block = (h,d), line stride 64, out += gamma*oW
    ccnl_axis_kernel<<<4096, 128, 0, stream>>>(q, k, v, m, z, x, gamma, out,
                                               /*strideA=*/64, /*M1=*/4096,
                                               /*M2=*/1, /*addX=*/0,
                                               /*maskDiag=*/0);
    // axis H: block = (w,d), line stride 4096, out += gamma*oH (diag masked)
    ccnl_axis_kernel<<<4096, 128, 0, stream>>>(q, k, v, m, z, x, gamma, out,
                                               /*strideA=*/4096, /*M1=*/64,
                                               /*M2=*/1, /*addX=*/0,
                                               /*maskDiag=*/1);
}